// RNN_25177098289484
// MI455X (gfx1250) — compile-verified
//
#include <hip/hip_runtime.h>
#include <hip/hip_bf16.h>

// S=10, B=16384, LAG=5, H=512, F=5
#define Sd 10
#define Bd 16384
#define Hd 512
#define Fd 5
#define G4 2048   // 4*H

typedef float v2f __attribute__((ext_vector_type(2)));
typedef float v8f __attribute__((ext_vector_type(8)));

__device__ __forceinline__ float sigmoidf_(float x) {
    return 1.0f / (1.0f + __expf(-x));
}

// ---------------------------------------------------------------------------
// Fold input projection + layer0 input gates into one tiny matrix:
//   Wc[j,l] = sum_h Wih0[j,h]*W_in[h,l]   (2048 x 5)
//   bc[j]   = sum_h Wih0[j,h]*b_in[h] + bih0[j]
// ---------------------------------------------------------------------------
__global__ __launch_bounds__(256) void build_wc_kernel(
    const float* __restrict__ Wih0, const float* __restrict__ W_in,
    const float* __restrict__ b_in, const float* __restrict__ bih0,
    float* __restrict__ Wc, float* __restrict__ bc)
{
    int j = blockIdx.x * blockDim.x + threadIdx.x;
    if (j >= G4) return;
    float acc[6] = {0.f, 0.f, 0.f, 0.f, 0.f, 0.f};
    const float* wrow = Wih0 + (size_t)j * Hd;
    for (int h = 0; h < Hd; ++h) {
        float w = wrow[h];
        #pragma unroll
        for (int l = 0; l < 5; ++l) acc[l] += w * W_in[h * 5 + l];
        acc[5] += w * b_in[h];
    }
    #pragma unroll
    for (int l = 0; l < 5; ++l) Wc[j * 5 + l] = acc[l];
    bc[j] = acc[5] + bih0[j];
}

__global__ __launch_bounds__(256) void zero_kernel(float* __restrict__ p, size_t n)
{
    size_t i = (size_t)blockIdx.x * blockDim.x + threadIdx.x;
    size_t stride = (size_t)gridDim.x * blockDim.x;
    for (; i < n; i += stride) p[i] = 0.f;
}

__global__ __launch_bounds__(256) void copy_kernel(
    float* __restrict__ dst, const float* __restrict__ src, size_t n)
{
    size_t i = (size_t)blockIdx.x * blockDim.x + threadIdx.x;
    size_t stride = (size_t)gridDim.x * blockDim.x;
    for (; i < n; i += stride) dst[i] = src[i];
}

// ---------------------------------------------------------------------------
// Layer 0 step: G = hprev @ Whh0^T (WMMA, K=512)  + x_t @ Wc^T + bc + bhh0,
// then LSTM pointwise; writes hnext and c in place.
// Block = 256 thr (8 waves). Patch: 64 batch rows x 32 hidden cols, x4 gates.
// Wave (8): 4x2 grid of 16x16 WMMA tiles; each wave keeps 4 gate accumulators.
// ---------------------------------------------------------------------------
__global__ __launch_bounds__(256) void lstm_l0_step_kernel(
    const float* __restrict__ hprev,   // [B,H]
    float* __restrict__ hnext,         // [B,H]
    float* __restrict__ c,             // [B,H] in/out
    const float* __restrict__ Whh,     // [4H,H] row-major
    const float* __restrict__ Wc,      // [4H,5]
    const float* __restrict__ bc,      // [4H]
    const float* __restrict__ bhh,     // [4H]
    const float* __restrict__ x,       // [S,B,5]
    int s)
{
    __shared__ float sh_h[64][68];     // A chunk: 64 rows x 64 K (+pad)
    __shared__ float sh_w[128][68];    // B chunk: 4 gates x 32 cols x 64 K
    __shared__ float sh_x[64][5];

    const int t = threadIdx.x;
    const int wave = t >> 5, lane = t & 31;
    const int lane_lo = lane & 15, half = lane >> 4;
    const int bBase = blockIdx.x * 64;
    const int nBase = blockIdx.y * 32;
    const int m0 = (wave >> 1) * 16;   // wave row offset in [0,64)
    const int nOff = (wave & 1) * 16;  // wave col offset in [0,32)
    const int n = nBase + nOff + lane_lo;   // this lane's hidden column

    // Per-lane gate-column constants (input-side folded gates + biases)
    float wcr[4][5];
    float bsum[4];
    #pragma unroll
    for (int g = 0; g < 4; ++g) {
        int gc = g * Hd + n;
        bsum[g] = bc[gc] + bhh[gc];
        #pragma unroll
        for (int l = 0; l < 5; ++l) wcr[g][l] = Wc[gc * 5 + l];
    }

    // Stage x tile (64 rows x 5)
    for (int i = t; i < 64 * 5; i += 256) {
        int r = i / 5, l = i % 5;
        sh_x[r][l] = x[((size_t)s * Bd + (bBase + r)) * 5 + l];
    }

    v8f acc[4];
    #pragma unroll
    for (int g = 0; g < 4; ++g) acc[g] = (v8f){0.f,0.f,0.f,0.f,0.f,0.f,0.f,0.f};

    for (int kb = 0; kb < Hd; kb += 64) {
        __syncthreads();
        // h chunk: 1024 float4
        #pragma unroll
        for (int i = 0; i < 4; ++i) {
            int q = t + 256 * i;
            int r = q >> 4, cf = q & 15;
            float4 v = *(const float4*)(hprev + (size_t)(bBase + r) * Hd + kb + cf * 4);
            *(float4*)&sh_h[r][cf * 4] = v;
        }
        // weight chunk: 2048 float4; row wr: gate = wr>>5, col-in-tile = wr&31
        #pragma unroll
        for (int i = 0; i < 8; ++i) {
            int q = t + 256 * i;
            int wr = q >> 4, cf = q & 15;
            int grow = (wr >> 5) * Hd + nBase + (wr & 31);
            float4 v = *(const float4*)(Whh + (size_t)grow * Hd + kb + cf * 4);
            *(float4*)&sh_w[wr][cf * 4] = v;
        }
        __syncthreads();
        #pragma unroll
        for (int kk = 0; kk < 64; kk += 4) {
            // A frag (16x4 f32): lane_lo row, K pair chosen by lane half
            v2f a = *(const v2f*)&sh_h[m0 + lane_lo][kk + 2 * half];
            #pragma unroll
            for (int g = 0; g < 4; ++g) {
                v2f b = *(const v2f*)&sh_w[g * 32 + nOff + lane_lo][kk + 2 * half];
                acc[g] = __builtin_amdgcn_wmma_f32_16x16x4_f32(
                    false, a, false, b, (short)0, acc[g], false, false);
            }
        }
    }

    // LSTM pointwise: C/D layout -> VGPR r holds M = r + 8*half, N = lane_lo
    #pragma unroll
    for (int r = 0; r < 8; ++r) {
        int mrow = m0 + r + 8 * half;
        int b_idx = bBase + mrow;
        float xv[5];
        #pragma unroll
        for (int l = 0; l < 5; ++l) xv[l] = sh_x[mrow][l];
        float gv[4];
        #pragma unroll
        for (int g = 0; g < 4; ++g) {
            float v = acc[g][r] + bsum[g];
            #pragma unroll
            for (int l = 0; l < 5; ++l) v += xv[l] * wcr[g][l];
            gv[g] = v;
        }
        size_t idx = (size_t)b_idx * Hd + n;
        float cp = c[idx];
        float ig = sigmoidf_(gv[0]);
        float fg = sigmoidf_(gv[1]);
        float gg = tanhf(gv[2]);
        float og = sigmoidf_(gv[3]);
        float cn = fg * cp + ig * gg;
        float hn = og * tanhf(cn);
        c[idx] = cn;
        hnext[idx] = hn;
    }
}

// ---------------------------------------------------------------------------
// Layer 1 step: G = h0new @ Wih1^T + h1prev @ Whh1^T + bih1 + bhh1,
// then LSTM pointwise; writes h1next and c1.
// ---------------------------------------------------------------------------
__global__ __launch_bounds__(256) void lstm_l1_step_kernel(
    const float* __restrict__ h0n,     // [B,H] layer-0 output (this step)
    const float* __restrict__ h1p,     // [B,H] layer-1 previous h
    float* __restrict__ h1n,           // [B,H]
    float* __restrict__ c,             // [B,H] in/out
    const float* __restrict__ Wih,     // [4H,H]
    const float* __restrict__ Whh,     // [4H,H]
    const float* __restrict__ bih,     // [4H]
    const float* __restrict__ bhh)     // [4H]
{
    __shared__ float sh_a0[64][68];
    __shared__ float sh_a1[64][68];
    __shared__ float sh_wa[128][68];
    __shared__ float sh_wb[128][68];

    const int t = threadIdx.x;
    const int wave = t >> 5, lane = t & 31;
    const int lane_lo = lane & 15, half = lane >> 4;
    const int bBase = blockIdx.x * 64;
    const int nBase = blockIdx.y * 32;
    const int m0 = (wave >> 1) * 16;
    const int nOff = (wave & 1) * 16;
    const int n = nBase + nOff + lane_lo;

    float bsum[4];
    #pragma unroll
    for (int g = 0; g < 4; ++g) {
        int gc = g * Hd + n;
        bsum[g] = bih[gc] + bhh[gc];
    }

    v8f acc[4];
    #pragma unroll
    for (int g = 0; g < 4; ++g) acc[g] = (v8f){0.f,0.f,0.f,0.f,0.f,0.f,0.f,0.f};

    for (int kb = 0; kb < Hd; kb += 64) {
        __syncthreads();
        #pragma unroll
        for (int i = 0; i < 4; ++i) {
            int q = t + 256 * i;
            int r = q >> 4, cf = q & 15;
            *(float4*)&sh_a0[r][cf * 4] =
                *(const float4*)(h0n + (size_t)(bBase + r) * Hd + kb + cf * 4);
            *(float4*)&sh_a1[r][cf * 4] =
                *(const float4*)(h1p + (size_t)(bBase + r) * Hd + kb + cf * 4);
        }
        #pragma unroll
        for (int i = 0; i < 8; ++i) {
            int q = t + 256 * i;
            int wr = q >> 4, cf = q & 15;
            int grow = (wr >> 5) * Hd + nBase + (wr & 31);
            *(float4*)&sh_wa[wr][cf * 4] =
                *(const float4*)(Wih + (size_t)grow * Hd + kb + cf * 4);
            *(float4*)&sh_wb[wr][cf * 4] =
                *(const float4*)(Whh + (size_t)grow * Hd + kb + cf * 4);
        }
        __syncthreads();
        #pragma unroll
        for (int kk = 0; kk < 64; kk += 4) {
            v2f a0 = *(const v2f*)&sh_a0[m0 + lane_lo][kk + 2 * half];
            v2f a1 = *(const v2f*)&sh_a1[m0 + lane_lo][kk + 2 * half];
            #pragma unroll
            for (int g = 0; g < 4; ++g) {
                v2f ba = *(const v2f*)&sh_wa[g * 32 + nOff + lane_lo][kk + 2 * half];
                acc[g] = __builtin_amdgcn_wmma_f32_16x16x4_f32(
                    false, a0, false, ba, (short)0, acc[g], false, false);
                v2f bb = *(const v2f*)&sh_wb[g * 32 + nOff + lane_lo][kk + 2 * half];
                acc[g] = __builtin_amdgcn_wmma_f32_16x16x4_f32(
                    false, a1, false, bb, (short)0, acc[g], false, false);
            }
        }
    }

    #pragma unroll
    for (int r = 0; r < 8; ++r) {
        int mrow = m0 + r + 8 * half;
        int b_idx = bBase + mrow;
        float gv[4];
        #pragma unroll
        for (int g = 0; g < 4; ++g) gv[g] = acc[g][r] + bsum[g];
        size_t idx = (size_t)b_idx * Hd + n;
        float cp = c[idx];
        float ig = sigmoidf_(gv[0]);
        float fg = sigmoidf_(gv[1]);
        float gg = tanhf(gv[2]);
        float og = sigmoidf_(gv[3]);
        float cn = fg * cp + ig * gg;
        float hn = og * tanhf(cn);
        c[idx] = cn;
        h1n[idx] = hn;
    }
}

// ---------------------------------------------------------------------------
// Per-step scalar head: out[b, s] = dot(h1[b,:], W_out) + b_out.
// One wave32 per batch row; shuffle-tree reduce.
// ---------------------------------------------------------------------------
__global__ __launch_bounds__(256) void step_out_kernel(
    const float* __restrict__ h1, const float* __restrict__ Wout,
    const float* __restrict__ bout, float* __restrict__ out, int s)
{
    int wave = threadIdx.x >> 5, lane = threadIdx.x & 31;
    int b = blockIdx.x * 8 + wave;
    float acc = 0.f;
    const float* hr = h1 + (size_t)b * Hd;
    for (int k = lane; k < Hd; k += 32) acc += hr[k] * Wout[k];
    #pragma unroll
    for (int off = 16; off > 0; off >>= 1) acc += __shfl_down(acc, off, 32);
    if (lane == 0) out[(size_t)b * (Sd + Fd) + s] = acc + bout[0];
}

// Forecast head: out[b, S+f] = dot(h1_final[b,:], W_f[f,:]) + b_f[f]
__global__ __launch_bounds__(256) void fc_kernel(
    const float* __restrict__ h1, const float* __restrict__ Wf,
    const float* __restrict__ bf, float* __restrict__ out)
{
    int wave = threadIdx.x >> 5, lane = threadIdx.x & 31;
    int b = blockIdx.x * 8 + wave;
    float acc[Fd] = {0.f, 0.f, 0.f, 0.f, 0.f};
    const float* hr = h1 + (size_t)b * Hd;
    for (int k = lane; k < Hd; k += 32) {
        float hv = hr[k];
        #pragma unroll
        for (int f = 0; f < Fd; ++f) acc[f] += hv * Wf[f * Hd + k];
    }
    #pragma unroll
    for (int f = 0; f < Fd; ++f) {
        #pragma unroll
        for (int off = 16; off > 0; off >>= 1) acc[f] += __shfl_down(acc[f], off, 32);
    }
    if (lane == 0) {
        #pragma unroll
        for (int f = 0; f < Fd; ++f)
            out[(size_t)b * (Sd + Fd) + Sd + f] = acc[f] + bf[f];
    }
}

// ---------------------------------------------------------------------------
extern "C" void kernel_launch(void* const* d_in, const int* in_sizes, int n_in,
                              void* d_out, int out_size, void* d_ws, size_t ws_size,
                              hipStream_t stream)
{
    const float* x     = (const float*)d_in[0];
    const float* W_in  = (const float*)d_in[1];
    const float* b_in  = (const float*)d_in[2];
    const float* Wih0  = (const float*)d_in[3];
    const float* Whh0  = (const float*)d_in[4];
    const float* bih0  = (const float*)d_in[5];
    const float* bhh0  = (const float*)d_in[6];
    const float* Wih1  = (const float*)d_in[7];
    const float* Whh1  = (const float*)d_in[8];
    const float* bih1  = (const float*)d_in[9];
    const float* bhh1  = (const float*)d_in[10];
    const float* W_out = (const float*)d_in[11];
    const float* b_out = (const float*)d_in[12];
    const float* W_f   = (const float*)d_in[13];
    const float* b_f   = (const float*)d_in[14];

    const size_t BH = (size_t)Bd * Hd;
    float* out = (float*)d_out;
    // d_out layout: outputs [B, S+F], then h_n [2,B,H], then c_n [2,B,H]
    float* out_h0 = out + (size_t)Bd * (Sd + Fd);
    float* out_h1 = out_h0 + BH;
    float* c0 = out_h1 + BH;     // c_n lives in d_out, updated in place
    float* c1 = c0 + BH;

    // Workspace: two ping-pong h-state sets + folded Wc/bc
    float* ws  = (float*)d_ws;
    float* hA0 = ws;
    float* hA1 = hA0 + BH;
    float* hB0 = hA1 + BH;
    float* hB1 = hB0 + BH;
    float* Wc  = hB1 + BH;
    float* bc  = Wc + (size_t)G4 * 5;

    // Zero initial h (buffer A) and both c states
    zero_kernel<<<1024, 256, 0, stream>>>(hA0, 2 * BH);
    zero_kernel<<<1024, 256, 0, stream>>>(c0, 2 * BH);

    build_wc_kernel<<<(G4 + 255) / 256, 256, 0, stream>>>(Wih0, W_in, b_in, bih0, Wc, bc);

    dim3 gemm_grid(Bd / 64, Hd / 32);   // 256 x 16 workgroups
    for (int s = 0; s < Sd; ++s) {
        float* hp0 = (s & 1) ? hB0 : hA0;
        float* hp1 = (s & 1) ? hB1 : hA1;
        float* hn0 = (s & 1) ? hA0 : hB0;
        float* hn1 = (s & 1) ? hA1 : hB1;
        lstm_l0_step_kernel<<<gemm_grid, 256, 0, stream>>>(
            hp0, hn0, c0, Whh0, Wc, bc, bhh0, x, s);
        lstm_l1_step_kernel<<<gemm_grid, 256, 0, stream>>>(
            hn0, hp1, hn1, c1, Wih1, Whh1, bih1, bhh1);
        step_out_kernel<<<Bd / 8, 256, 0, stream>>>(hn1, W_out, b_out, out, s);
    }
    // After s=9 (odd), final states are in buffer A (hA0, hA1 contiguous)
    copy_kernel<<<1024, 256, 0, stream>>>(out_h0, hA0, 2 * BH);
    fc_kernel<<<Bd / 8, 256, 0, stream>>>(hA1, W_f, b_f, out);
}